// TopKRouterCommunication_80015240724742
// MI455X (gfx1250) — compile-verified
//
#include <hip/hip_runtime.h>
#include <hip/hip_bf16.h>
#include <math.h>

// ---------------------------------------------------------------------------
// TopK router communication, MI455X (gfx1250, wave32).
//
// Shapes (fixed by reference setup): B=8, N=4096, D=1024, TOPK=2.
// Key optimization: the reference computes the full router GEMM
// [B*N,D]x[D,D] (~68.7 GFLOP) but only uses B*TOPK = 16 rows of it.
// We compute scores (bandwidth pass), pick top-2 tokens per batch,
// run a 16xD @ DxD GEMM via V_WMMA_F32_16X16X4_F32 (M=16 == one WMMA
// tile), and fuse add+LayerNorm in a final bandwidth pass.
// ---------------------------------------------------------------------------

typedef float v2f __attribute__((ext_vector_type(2)));
typedef float v8f __attribute__((ext_vector_type(8)));

#define D_MODEL 1024
#define N_TOK   4096

// ---------------------------------------------------------------------------
// Kernel 1: scores[b,n] = dot(x[b,n,:], w_score)  (masked by alive).
// One wave per token; lane handles 8 float4 chunks (coalesced 512B/step).
// b_score is a constant shift -> irrelevant for top-k, skipped.
// ---------------------------------------------------------------------------
__global__ void k_scores(const float* __restrict__ x,
                         const unsigned char* __restrict__ alive,
                         const float* __restrict__ w_score,
                         float* __restrict__ scores, int BN) {
  const int wave = (blockIdx.x * blockDim.x + threadIdx.x) >> 5;
  const int lane = threadIdx.x & 31;
  if (wave >= BN) return;
  const float4* xr = (const float4*)(x + (size_t)wave * D_MODEL);
  const float4* wr = (const float4*)w_score;
  float acc = 0.f;
#pragma unroll
  for (int i = 0; i < 8; ++i) {
    const int idx = i * 32 + lane;          // 256 float4 = 1024 floats
    const float4 a = xr[idx];
    const float4 w = wr[idx];
    acc += a.x * w.x + a.y * w.y + a.z * w.z + a.w * w.w;
  }
#pragma unroll
  for (int o = 16; o > 0; o >>= 1) acc += __shfl_xor(acc, o, 32);
  if (lane == 0)
    scores[wave] = alive[wave] ? acc : -__builtin_inff();
}

// ---------------------------------------------------------------------------
// Kernel 2: per-batch top-2 (jax.lax.top_k tie rule: larger value wins;
// on equal value the smaller index wins). One block per batch.
// ---------------------------------------------------------------------------
__device__ __forceinline__ bool rank_gt(float v1, int i1, float v2, int i2) {
  return (v1 > v2) || (v1 == v2 && i1 < i2);
}

__global__ void k_top2(const float* __restrict__ scores,
                       int* __restrict__ topidx, int N) {
  const int b = blockIdx.x;
  const float* s = scores + (size_t)b * N;
  float bv = -__builtin_inff(), sv = -__builtin_inff();
  int bi = 0x7fffffff, si = 0x7fffffff;
  for (int i = threadIdx.x; i < N; i += blockDim.x) {
    const float v = s[i];
    if (rank_gt(v, i, bv, bi)) { sv = bv; si = bi; bv = v; bi = i; }
    else if (rank_gt(v, i, sv, si)) { sv = v; si = i; }
  }
  __shared__ float shv[512];
  __shared__ int   shi[512];
  shv[threadIdx.x] = bv;        shi[threadIdx.x] = bi;
  shv[256 + threadIdx.x] = sv;  shi[256 + threadIdx.x] = si;
  __syncthreads();
  for (int stride = 128; stride > 0; stride >>= 1) {
    if (threadIdx.x < stride) {
      float b1 = shv[threadIdx.x],          s1 = shv[256 + threadIdx.x];
      int   i1 = shi[threadIdx.x],          j1 = shi[256 + threadIdx.x];
      float b2 = shv[threadIdx.x + stride], s2 = shv[256 + threadIdx.x + stride];
      int   i2 = shi[threadIdx.x + stride], j2 = shi[256 + threadIdx.x + stride];
      float nb, ns; int nbi, nsi;
      if (rank_gt(b1, i1, b2, i2)) {
        nb = b1; nbi = i1;
        if (rank_gt(b2, i2, s1, j1)) { ns = b2; nsi = i2; }
        else                         { ns = s1; nsi = j1; }
      } else {
        nb = b2; nbi = i2;
        if (rank_gt(b1, i1, s2, j2)) { ns = b1; nsi = i1; }
        else                         { ns = s2; nsi = j2; }
      }
      shv[threadIdx.x] = nb;       shi[threadIdx.x] = nbi;
      shv[256 + threadIdx.x] = ns; shi[256 + threadIdx.x] = nsi;
    }
    __syncthreads();
  }
  if (threadIdx.x == 0) {
    topidx[b * 2 + 0] = shi[0];
    topidx[b * 2 + 1] = shi[256];
  }
}

// ---------------------------------------------------------------------------
// Kernel 3: gathered-row GEMM via WMMA fp32 + fused mean-over-k + bias.
//   A = 16 gathered x rows (B*TOPK = 16 = one WMMA M tile), staged in LDS
//       transposed As[k][m] (conflict-free A-fragment reads: addr = k*16+m).
//   B = W_router (DxD row-major), streamed from global (L2-resident, 4MB).
//   D = A x B accumulated over K=1024 in 256 v_wmma_f32_16x16x4_f32 steps.
// Epilogue: summary[b][col] = 0.5*(routed[2b]+routed[2b+1]) + b_router[col].
// Grid: 8 blocks x 8 waves; wave handles one 16-wide N tile (64 tiles).
//
// WMMA 16x16x4 fp32 lane layout (ISA 7.12.2):
//   A frag: lane<16 -> {A[m][k+0], A[m][k+1]}, lane>=16 -> {A[m][k+2], A[m][k+3]}
//   B frag: lane<16 -> {B[k+0][n], B[k+1][n]}, lane>=16 -> {B[k+2][n], B[k+3][n]}
//   C/D:    vgpr r: lane<16 -> D[r][n], lane>=16 -> D[r+8][n]
// ---------------------------------------------------------------------------
__global__ void k_router_wmma(const float* __restrict__ x,
                              const float* __restrict__ W,
                              const float* __restrict__ b_router,
                              const int* __restrict__ topidx,
                              float* __restrict__ summary, int N) {
  __shared__ float As[D_MODEL][16];          // [k][m], 64 KB, transposed

  // Cooperative gather of the 16 selected rows into LDS (transposed).
  for (int q = threadIdx.x; q < 16 * (D_MODEL / 4); q += blockDim.x) {
    const int row = q >> 8;                  // q / 256
    const int c4  = q & 255;
    const int b   = row >> 1;
    const int sel = row & 1;
    const int tok = topidx[b * 2 + sel];
    const float4 v =
        ((const float4*)(x + ((size_t)b * N + tok) * D_MODEL))[c4];
    const int c = c4 * 4;
    As[c + 0][row] = v.x; As[c + 1][row] = v.y;
    As[c + 2][row] = v.z; As[c + 3][row] = v.w;
  }
  __syncthreads();

  const int wave  = threadIdx.x >> 5;
  const int lane  = threadIdx.x & 31;
  const int nt    = blockIdx.x * 8 + wave;   // 0..63
  const int n0    = nt * 16;
  const int m     = lane & 15;
  const int khalf = lane >> 4;               // 0 or 1
  const int ncol  = n0 + m;

  v8f c = {};
  for (int k = 0; k < D_MODEL; k += 4) {
    const int kk = k + khalf * 2;
    v2f a, bf;
    a.x  = As[kk][m];
    a.y  = As[kk + 1][m];
    bf.x = W[(size_t)kk * D_MODEL + ncol];
    bf.y = W[(size_t)(kk + 1) * D_MODEL + ncol];
    c = __builtin_amdgcn_wmma_f32_16x16x4_f32(
        /*neg_a=*/false, a, /*neg_b=*/false, bf,
        /*c_mod=*/(short)0, c, /*reuse_a=*/false, /*reuse_b=*/false);
  }

  // c[r] = routed[r + 8*khalf][ncol]; rows (2b, 2b+1) -> batch b.
  const float bias = b_router[ncol];
#pragma unroll
  for (int j = 0; j < 4; ++j) {
    const int b = 4 * khalf + j;
    const float s = 0.5f * (c[2 * j] + c[2 * j + 1]) + bias;
    summary[(size_t)b * D_MODEL + ncol] = s;
  }
}

// ---------------------------------------------------------------------------
// Kernel 4: out = LayerNorm(x + summary[b]) * gamma + beta.
// One wave per token; 32 floats/lane in registers; single-pass sum/sumsq.
// ---------------------------------------------------------------------------
__global__ void k_add_ln(const float* __restrict__ x,
                         const float* __restrict__ summary,
                         const float* __restrict__ gamma,
                         const float* __restrict__ beta,
                         float* __restrict__ out, int N) {
  const int wave = (blockIdx.x * blockDim.x + threadIdx.x) >> 5;
  const int lane = threadIdx.x & 31;
  const int b = wave / N;
  const float4* xr = (const float4*)(x + (size_t)wave * D_MODEL);
  const float4* sr = (const float4*)(summary + (size_t)b * D_MODEL);

  float4 r[8];
  float sum = 0.f, sq = 0.f;
#pragma unroll
  for (int i = 0; i < 8; ++i) {
    const int idx = i * 32 + lane;
    const float4 xv = xr[idx];
    const float4 sv = sr[idx];
    float4 y;
    y.x = xv.x + sv.x; y.y = xv.y + sv.y;
    y.z = xv.z + sv.z; y.w = xv.w + sv.w;
    r[i] = y;
    sum += y.x + y.y + y.z + y.w;
    sq  += y.x * y.x + y.y * y.y + y.z * y.z + y.w * y.w;
  }
#pragma unroll
  for (int o = 16; o > 0; o >>= 1) {
    sum += __shfl_xor(sum, o, 32);
    sq  += __shfl_xor(sq,  o, 32);
  }
  const float inv_d = 1.0f / (float)D_MODEL;
  const float mu    = sum * inv_d;
  const float var   = sq * inv_d - mu * mu;
  const float rstd  = rsqrtf(var + 1e-5f);

  const float4* gr = (const float4*)gamma;
  const float4* br = (const float4*)beta;
  float4* orow = (float4*)(out + (size_t)wave * D_MODEL);
#pragma unroll
  for (int i = 0; i < 8; ++i) {
    const int idx = i * 32 + lane;
    const float4 g = gr[idx];
    const float4 be = br[idx];
    const float4 y = r[i];
    float4 o4;
    o4.x = (y.x - mu) * rstd * g.x + be.x;
    o4.y = (y.y - mu) * rstd * g.y + be.y;
    o4.z = (y.z - mu) * rstd * g.z + be.z;
    o4.w = (y.w - mu) * rstd * g.w + be.w;
    orow[idx] = o4;
  }
}

// ---------------------------------------------------------------------------
// Launch. Inputs (setup_inputs order):
//   0:x [B,N,D] f32, 1:alive_mask [B,N] bool(u8), 2:W_router [D,D] f32,
//   3:b_router [D] f32, 4:w_score [D] f32, 5:b_score [1] f32,
//   6:gamma [D] f32, 7:beta [D] f32.  Output: [B,N,D] f32.
// ---------------------------------------------------------------------------
extern "C" void kernel_launch(void* const* d_in, const int* in_sizes, int n_in,
                              void* d_out, int out_size, void* d_ws,
                              size_t ws_size, hipStream_t stream) {
  const float* x              = (const float*)d_in[0];
  const unsigned char* alive  = (const unsigned char*)d_in[1];
  const float* W_router       = (const float*)d_in[2];
  const float* b_router       = (const float*)d_in[3];
  const float* w_score        = (const float*)d_in[4];
  const float* gamma          = (const float*)d_in[6];
  const float* beta           = (const float*)d_in[7];
  float* out                  = (float*)d_out;

  const int N  = N_TOK;
  const int B  = in_sizes[1] / N;   // alive_mask has B*N elements (B=8)
  const int BN = B * N;

  // Workspace: scores[BN] f32 | topidx[B*2] i32 (256B slot) | summary[B*D] f32
  float* scores  = (float*)d_ws;
  int*   topidx  = (int*)((char*)d_ws + (size_t)BN * sizeof(float));
  float* summary = (float*)((char*)d_ws + (size_t)BN * sizeof(float) + 256);

  // 1) scores: one wave per token (8 waves / 256-thread block).
  k_scores<<<BN / 8, 256, 0, stream>>>(x, alive, w_score, scores, BN);
  // 2) top-2 token indices per batch.
  k_top2<<<B, 256, 0, stream>>>(scores, topidx, N);
  // 3) 16-row router GEMM via WMMA fp32 + fused mean + bias -> summary.
  k_router_wmma<<<(D_MODEL / 16) / 8, 256, 0, stream>>>(
      x, W_router, b_router, topidx, summary, N);
  // 4) fused residual-add + LayerNorm.
  k_add_ln<<<BN / 8, 256, 0, stream>>>(x, summary, gamma, beta, out, N);
}